// WavKANLayer_46334107189759
// MI455X (gfx1250) — compile-verified
//
#include <hip/hip_runtime.h>
#include <math.h>

#define B_DIM   1024
#define OUT_DIM 512
#define IN_DIM  512
#define EPS_BN  1e-5f
// 2 / (sqrt(3) * pi^0.25)
#define MH_C    0.86732507058407749f

typedef __attribute__((ext_vector_type(2))) float v2f;
typedef __attribute__((ext_vector_type(8))) float v8f;

// ---------------------------------------------------------------------------
// Kernel 1: sx = silu(x), vectorized 4-wide (global_load_b128 / store_b128)
// ---------------------------------------------------------------------------
__global__ __launch_bounds__(256) void silu_kernel(const float* __restrict__ x,
                                                   float* __restrict__ sx) {
    int idx = (blockIdx.x * 256 + threadIdx.x) * 4;
    float4 v = *(const float4*)(x + idx);
    float4 r;
    r.x = v.x / (1.0f + __expf(-v.x));
    r.y = v.y / (1.0f + __expf(-v.y));
    r.z = v.z / (1.0f + __expf(-v.z));
    r.w = v.w / (1.0f + __expf(-v.w));
    *(float4*)(sx + idx) = r;
}

// ---------------------------------------------------------------------------
// Kernel 2: y = sx @ w1^T using V_WMMA_F32_16X16X4_F32 (full fp32 precision).
// One wave owns one 16x16 output tile; 8 waves/block cover a 32(b) x 64(o)
// block tile. K=512 -> 128 chained WMMAs per wave.
// ---------------------------------------------------------------------------
__global__ __launch_bounds__(256) void base_gemm_kernel(const float* __restrict__ sx,
                                                        const float* __restrict__ w1,
                                                        float* __restrict__ y) {
    const int lane = threadIdx.x & 31;
    const int wave = threadIdx.x >> 5;      // 0..7
    const int wb   = wave >> 2;             // 0..1  (b sub-tile)
    const int wo   = wave & 3;              // 0..3  (o sub-tile)
    const int b0   = blockIdx.y * 32 + wb * 16;
    const int o0   = blockIdx.x * 64 + wo * 16;

    const int mn   = lane & 15;             // A-row / B-col owned by this lane
    const int koff = (lane >> 4) * 2;       // K sub-offset per half-wave

    // A: sx is (B, IN) row-major -> lane reads sx[b0+mn][kk+koff .. +1]
    const float* aptr = sx + (size_t)(b0 + mn) * IN_DIM + koff;
    // B: w1 is (OUT, IN) row-major, B[k][n] = w1[o0+n][k]
    const float* bptr = w1 + (size_t)(o0 + mn) * IN_DIM + koff;

    v8f acc = {};
    #pragma unroll 4
    for (int kk = 0; kk < IN_DIM; kk += 4) {
        v2f a = *(const v2f*)(aptr + kk);
        v2f b = *(const v2f*)(bptr + kk);
        acc = __builtin_amdgcn_wmma_f32_16x16x4_f32(
            /*neg_a=*/false, a, /*neg_b=*/false, b,
            /*c_mod=*/(short)0, acc, /*reuse_a=*/false, /*reuse_b=*/false);
    }

    // D layout: VGPR j -> M = j + (lane>>4)*8, N = lane&15
    const int mrow = (lane >> 4) * 8;
    #pragma unroll
    for (int j = 0; j < 8; ++j)
        y[(size_t)(b0 + mrow + j) * OUT_DIM + (o0 + mn)] = acc[j];
}

// ---------------------------------------------------------------------------
// Kernel 3: y += einsum('boi,oi->bo', mexican_hat((x-t)/s), ww)
// LDS-tiled: block = 16(b) x 16(o), i processed in chunks of 16.
// This is the compute-dominant path: ~268M exp evaluations, VALU-bound.
// ---------------------------------------------------------------------------
__global__ __launch_bounds__(256) void wavelet_kernel(const float* __restrict__ x,
                                                      const float* __restrict__ scl,
                                                      const float* __restrict__ trn,
                                                      const float* __restrict__ ww,
                                                      float* __restrict__ y) {
    __shared__ float xs[16][17];   // [b][i]
    __shared__ float ts[16][17];   // [o][i]
    __shared__ float ss[16][17];   // [o][i]
    __shared__ float wv[16][17];   // [o][i]

    const int tx = threadIdx.x;    // o within tile
    const int ty = threadIdx.y;    // b within tile
    const int o0 = blockIdx.x * 16;
    const int b0 = blockIdx.y * 16;

    float acc = 0.0f;
    for (int ic = 0; ic < IN_DIM; ic += 16) {
        // cooperative tile loads (ty reused as row index for both b- and o-tiles)
        xs[ty][tx] = x  [(size_t)(b0 + ty) * IN_DIM + ic + tx];
        ts[ty][tx] = trn[(size_t)(o0 + ty) * IN_DIM + ic + tx];
        ss[ty][tx] = scl[(size_t)(o0 + ty) * IN_DIM + ic + tx];
        wv[ty][tx] = ww [(size_t)(o0 + ty) * IN_DIM + ic + tx];
        __syncthreads();
        #pragma unroll
        for (int k = 0; k < 16; ++k) {
            float z  = (xs[ty][k] - ts[tx][k]) / ss[tx][k];
            float z2 = z * z;
            acc += (z2 - 1.0f) * __expf(-0.5f * z2) * wv[tx][k];
        }
        __syncthreads();
    }
    y[(size_t)(b0 + ty) * OUT_DIM + (o0 + tx)] += MH_C * acc;
}

// ---------------------------------------------------------------------------
// Kernel 4: BatchNorm1d (training mode, biased variance) over the batch dim.
// One block per output channel o.
// ---------------------------------------------------------------------------
__global__ __launch_bounds__(256) void bn_kernel(const float* __restrict__ y,
                                                 const float* __restrict__ gamma,
                                                 const float* __restrict__ beta,
                                                 float* __restrict__ out) {
    __shared__ float s1[256];
    __shared__ float s2[256];
    const int o = blockIdx.x;
    const int t = threadIdx.x;

    float sum = 0.0f, sq = 0.0f;
    for (int b = t; b < B_DIM; b += 256) {
        float v = y[(size_t)b * OUT_DIM + o];
        sum += v;
        sq  += v * v;
    }
    s1[t] = sum;
    s2[t] = sq;
    __syncthreads();
    for (int s = 128; s > 0; s >>= 1) {
        if (t < s) { s1[t] += s1[t + s]; s2[t] += s2[t + s]; }
        __syncthreads();
    }
    const float mean = s1[0] * (1.0f / B_DIM);
    const float var  = s2[0] * (1.0f / B_DIM) - mean * mean;
    const float g    = rsqrtf(var + EPS_BN) * gamma[o];
    const float bo   = beta[o];
    for (int b = t; b < B_DIM; b += 256) {
        out[(size_t)b * OUT_DIM + o] = (y[(size_t)b * OUT_DIM + o] - mean) * g + bo;
    }
}

// ---------------------------------------------------------------------------
extern "C" void kernel_launch(void* const* d_in, const int* in_sizes, int n_in,
                              void* d_out, int out_size, void* d_ws, size_t ws_size,
                              hipStream_t stream) {
    const float* x     = (const float*)d_in[0];
    const float* scl   = (const float*)d_in[1];
    const float* trn   = (const float*)d_in[2];
    const float* w1    = (const float*)d_in[3];
    const float* ww    = (const float*)d_in[4];
    const float* gamma = (const float*)d_in[5];
    const float* beta  = (const float*)d_in[6];
    float* out = (float*)d_out;

    float* sx = (float*)d_ws;                 // B*IN  floats (2 MB)
    float* y  = sx + (size_t)B_DIM * IN_DIM;  // B*OUT floats (2 MB)

    // 1) sx = silu(x)
    silu_kernel<<<(B_DIM * IN_DIM) / (256 * 4), 256, 0, stream>>>(x, sx);

    // 2) y = sx @ w1^T  (WMMA f32)
    base_gemm_kernel<<<dim3(OUT_DIM / 64, B_DIM / 32), 256, 0, stream>>>(sx, w1, y);

    // 3) y += wavelet path
    wavelet_kernel<<<dim3(OUT_DIM / 16, B_DIM / 16), dim3(16, 16), 0, stream>>>(
        x, scl, trn, ww, y);

    // 4) out = batchnorm(y) * gamma + beta
    bn_kernel<<<OUT_DIM, 256, 0, stream>>>(y, gamma, beta, out);
}